// SiamRPNTHORForward_21646635172259
// MI455X (gfx1250) — compile-verified
//
#include <hip/hip_runtime.h>
#include <math.h>

typedef __attribute__((ext_vector_type(2))) float v2f;
typedef __attribute__((ext_vector_type(8))) float v8f;

// ---------------- workspace layout (float offsets) ----------------
// XP   : NCHW input 3*287*287 = 247,107            -> 247,168
// APAD : padded weights, max 384*3456 = 1,327,104  -> 1,331,200
// ACT0 : activation ping, max 96*19321 = 1,854,816 -> 1,900,544
// ACT1 : activation pong                           -> 1,900,544
// COL  : BT im2col, max 4288*2400 = 10,291,200     -> 10,297,344
#define WS_XP    0
#define WS_APAD  247168
#define WS_ACT0  (247168 + 1331200)
#define WS_ACT1  (WS_ACT0 + 1900544)
#define WS_COL   (WS_ACT1 + 1900544)
// aliases inside COL region (backbone col dead by then)
#define WS_CP    WS_COL                    // unfold of c_x : 256*16*441 = 1,806,336
#define WS_RP    (WS_COL + 1835008)        // unfold of r_x
#define WS_CCOR  (WS_COL + 3670016)        // cls corr BT (3584 x 256) = 917,504
#define WS_RCOR  (WS_COL + 4587520)        // reg corr BT
// aliases in ACT regions for heads
#define WS_H1T   WS_ACT0                   // head hidden BT (3584 x 256) = 917,504
#define WS_CLS   WS_ACT1                   // (10,3528) = 35,280
#define WS_REG   (WS_ACT1 + 131072)        // (20,3528) = 70,560

// ---------------- NHWC -> NCHW (287x287x3) ----------------
__global__ void k_nhwc_to_nchw(const float* __restrict__ x, float* __restrict__ xp,
                               int H, int W, int C) {
  int idx = blockIdx.x * blockDim.x + threadIdx.x;
  int total = H * W * C;
  if (idx >= total) return;
  int c = idx % C;
  int hw = idx / C;
  xp[c * H * W + hw] = x[idx];
}

// ------- weight pad: W[MxK] row-major -> AP[M16xK4] zero-padded -------
__global__ void k_padA(const float* __restrict__ W, float* __restrict__ AP,
                       int M, int K, int M16, int K4) {
  long long idx = (long long)blockIdx.x * blockDim.x + threadIdx.x;
  long long total = (long long)M16 * K4;
  if (idx >= total) return;
  int k = (int)(idx % K4);
  int m = (int)(idx / K4);
  AP[idx] = (m < M && k < K) ? W[(long long)m * K + k] : 0.0f;
}

// ------- im2col transposed+padded: in (C,H,W) -> BT[NPAD][K4] (zero-filled) -------
__global__ void k_im2colT(const float* __restrict__ in, float* __restrict__ bt,
                          int C, int H, int W, int KH, int KW, int S,
                          int OH, int OW, int NPAD, int K, int K4) {
  long long idx = (long long)blockIdx.x * blockDim.x + threadIdx.x;
  long long total = (long long)NPAD * K4;
  if (idx >= total) return;
  int k = (int)(idx % K4);
  int n = (int)(idx / K4);
  int N = OH * OW;
  float v = 0.0f;
  if (n < N && k < K) {
    int kw = k % KW; int t = k / KW;
    int kh = t % KH; int c  = t / KH;
    int ox = n % OW, oy = n / OW;
    v = in[((long long)c * H + (oy * S + kh)) * W + (ox * S + kw)];
  }
  bt[idx] = v;
}

// ------------- fp32 WMMA GEMM, padded operands, pipelined K loop -------------
// A  : [M16 x K4] row-major (zero-padded weights)
// BT : [NPAD x K4] row-major, NPAD multiple of 64 (zero-padded columns)
// Wave tile = 16(M) x 64(N). Per K-chunk: 1 A b64 + 4 B b64 + 4 WMMA.
// 3-stage register pipeline keeps up to 15 loads in flight so waits become
// partial (s_wait_loadcnt != 0) and WMMAs overlap memory latency.
// storeT==0: C[M x N] row-major. storeT==1: CT[NPAD x M] (next GEMM's BT).
__global__ void __launch_bounds__(256)
k_gemm_wmma(const float* __restrict__ A, const float* __restrict__ BT,
            const float* __restrict__ bias, float* __restrict__ C,
            int M, int M16, int N, int NPAD, int K4, int relu, int storeT) {
  const int lane  = threadIdx.x & 31;
  const int wave  = threadIdx.x >> 5;
  const int gwave = blockIdx.x * (blockDim.x >> 5) + wave;
  const int ntiles = NPAD >> 6;
  const int mtiles = M16 >> 4;
  if (gwave >= ntiles * mtiles) return;           // wave-uniform
  const int tm = gwave / ntiles;
  const int tn = gwave - tm * ntiles;
  const int m0 = tm << 4;
  const int n0 = tn << 6;

  const int l15   = lane & 15;
  const int khalf = (lane >> 4) << 1;             // 0 or 2

  const float* ap = A  + (size_t)(m0 + l15) * K4 + khalf;
  const float* bp = BT + (size_t)(n0 + l15) * K4 + khalf;
  const size_t bs = (size_t)16 * K4;              // 16 columns ahead in BT

  v8f acc0 = {}, acc1 = {}, acc2 = {}, acc3 = {};

  struct Frag { v2f a, b0, b1, b2, b3; };
  auto ld = [&](int off) {
    Frag f;
    f.a  = *(const v2f*)(ap + off);
    f.b0 = *(const v2f*)(bp + off);
    f.b1 = *(const v2f*)(bp + bs + off);
    f.b2 = *(const v2f*)(bp + 2 * bs + off);
    f.b3 = *(const v2f*)(bp + 3 * bs + off);
    return f;
  };
  auto mm = [&](const Frag& f) {
    acc0 = __builtin_amdgcn_wmma_f32_16x16x4_f32(false, f.a, false, f.b0, (short)0, acc0, false, false);
    acc1 = __builtin_amdgcn_wmma_f32_16x16x4_f32(false, f.a, false, f.b1, (short)0, acc1, false, false);
    acc2 = __builtin_amdgcn_wmma_f32_16x16x4_f32(false, f.a, false, f.b2, (short)0, acc2, false, false);
    acc3 = __builtin_amdgcn_wmma_f32_16x16x4_f32(false, f.a, false, f.b3, (short)0, acc3, false, false);
  };

  if (K4 >= 12) {
    Frag f0 = ld(0);
    Frag f1 = ld(4);
    Frag f2 = ld(8);
    for (int k0 = 12; k0 < K4; k0 += 4) {
      Frag fn = ld(k0);       // issue next chunk's loads before consuming f0
      mm(f0);
      f0 = f1; f1 = f2; f2 = fn;
    }
    mm(f0);
    mm(f1);
    mm(f2);
  } else {
    for (int k0 = 0; k0 < K4; k0 += 4) {
      Frag f = ld(k0);
      mm(f);
    }
  }

  // C/D layout: VGPR r -> row m0+r (lanes 0-15) or m0+8+r (lanes 16-31); col = +(lane&15)
  const int mBase = m0 + ((lane >> 4) << 3);
  if (storeT == 0) {
#pragma unroll
    for (int j = 0; j < 4; ++j) {
      v8f acc = (j == 0) ? acc0 : (j == 1) ? acc1 : (j == 2) ? acc2 : acc3;
      int bc = n0 + j * 16 + l15;
      if (bc < N) {
#pragma unroll
        for (int r = 0; r < 8; ++r) {
          int mm2 = mBase + r;
          if (mm2 < M) {
            float v = acc[r] + bias[mm2];
            if (relu) v = fmaxf(v, 0.0f);
            C[(long long)mm2 * N + bc] = v;
          }
        }
      }
    }
  } else {
    // transposed store into next GEMM's BT layout [NPAD x M]; pad rows -> 0
#pragma unroll
    for (int j = 0; j < 4; ++j) {
      v8f acc = (j == 0) ? acc0 : (j == 1) ? acc1 : (j == 2) ? acc2 : acc3;
      int bc = n0 + j * 16 + l15;
      bool ok = bc < N;
#pragma unroll
      for (int r = 0; r < 8; ++r) {
        int mm2 = mBase + r;
        if (mm2 < M) {
          float v = ok ? (acc[r] + bias[mm2]) : 0.0f;
          if (relu) v = fmaxf(v, 0.0f);
          C[(long long)bc * M + mm2] = v;
        }
      }
    }
  }
}

// ---------------- 3x3 stride-2 maxpool + relu ----------------
__global__ void k_maxpool3s2_relu(const float* __restrict__ in, float* __restrict__ out,
                                  int C, int H, int W, int OH, int OW) {
  int idx = blockIdx.x * blockDim.x + threadIdx.x;
  int total = C * OH * OW;
  if (idx >= total) return;
  int ox = idx % OW; int t = idx / OW;
  int oy = t % OH;  int c = t / OH;
  const float* p = in + (long long)c * H * W;
  int y0 = oy * 2, x0 = ox * 2;
  float m = -INFINITY;
#pragma unroll
  for (int dy = 0; dy < 3; ++dy)
#pragma unroll
    for (int dx = 0; dx < 3; ++dx)
      m = fmaxf(m, p[(y0 + dy) * W + (x0 + dx)]);
  out[idx] = fmaxf(m, 0.0f);   // relu(maxpool(x))
}

// ---------------- unfold 4x4: (256,24,24) -> (256,16,441) ----------------
__global__ void k_unfold44(const float* __restrict__ in, float* __restrict__ out) {
  int idx = blockIdx.x * blockDim.x + threadIdx.x;
  const int total = 256 * 16 * 441;
  if (idx >= total) return;
  int l = idx % 441; int t = idx / 441;
  int k = t % 16;    int c = t / 16;
  int kh = k >> 2, kw = k & 3;
  int oy = l / 21, ox = l % 21;
  out[idx] = in[((long long)c * 24 + (oy + kh)) * 24 + (ox + kw)];
}

// ------- depthwise xcorr -> head-GEMM BT layout [3584][256], pad rows = 0 -------
// bt[n][c] = sum_k p[c][k][l] * z[m][c][k], n = m*441 + l
__global__ void k_xcorrT(const float* __restrict__ p, const float* __restrict__ z,
                         float* __restrict__ bt) {
  int idx = blockIdx.x * blockDim.x + threadIdx.x;
  const int total = 3584 * 256;
  if (idx >= total) return;
  int c = idx & 255;
  int n = idx >> 8;
  float s = 0.0f;
  if (n < 3528) {
    int m = n / 441, l = n % 441;
    const float* pc = p + ((long long)c * 16) * 441 + l;
    const float* zc = z + ((long long)m * 256 + c) * 16;
#pragma unroll
    for (int k = 0; k < 16; ++k) s += pc[k * 441] * zc[k];
  }
  bt[idx] = s;
}

// ---------------- softmax + anchor decode ----------------
__global__ void k_decode(const float* __restrict__ cls_t, const float* __restrict__ reg_t,
                         const float* __restrict__ anc0, const float* __restrict__ anc1,
                         const float* __restrict__ anc2, const float* __restrict__ anc3,
                         float* __restrict__ out) {
  int idx = blockIdx.x * blockDim.x + threadIdx.x;
  const int total = 8 * 2205;
  if (idx >= total) return;
  int j = idx % 2205; int m = idx / 2205;
  int c5 = j / 441;   int l = j % 441;
  int col = m * 441 + l;

  float a  = cls_t[(c5)     * 3528 + col];
  float b  = cls_t[(c5 + 5) * 3528 + col];
  float mx = fmaxf(a, b);
  float ea = expf(a - mx), eb = expf(b - mx);
  float score1 = eb / (ea + eb);

  float dl0 = reg_t[(c5)      * 3528 + col];
  float dl1 = reg_t[(c5 + 5)  * 3528 + col];
  float dl2 = reg_t[(c5 + 10) * 3528 + col];
  float dl3 = reg_t[(c5 + 15) * 3528 + col];

  float A0 = anc0[idx], A1 = anc1[idx], A2 = anc2[idx], A3 = anc3[idx];
  const int STR = 8 * 2205;
  out[0 * STR + idx] = dl0 * A2 + A0;
  out[1 * STR + idx] = dl1 * A3 + A1;
  out[2 * STR + idx] = expf(dl2) * A2;
  out[3 * STR + idx] = expf(dl3) * A3;
  out[4 * STR + idx] = score1;
}

// ---------------- host-side helpers ----------------
static inline int cdiv(int a, int b) { return (a + b - 1) / b; }
static inline int pad64(int n) { return cdiv(n, 64) * 64; }

static void gemm(const float* W, const float* BT, const float* bias, float* C,
                 int M, int N, int NPAD, int K, float* APAD,
                 int relu, int storeT, hipStream_t s) {
  int M16 = cdiv(M, 16) * 16;
  int K4  = cdiv(K, 4) * 4;
  long long padTot = (long long)M16 * K4;
  hipLaunchKernelGGL(k_padA, dim3((int)((padTot + 255) / 256)), dim3(256), 0, s,
                     W, APAD, M, K, M16, K4);
  int waves = (M16 / 16) * (NPAD / 64);
  hipLaunchKernelGGL(k_gemm_wmma, dim3(cdiv(waves, 8)), dim3(256), 0, s,
                     APAD, BT, bias, C, M, M16, N, NPAD, K4, relu, storeT);
}

static void im2colT(const float* in, float* bt, int C, int H, int W,
                    int KH, int KW, int S, int OH, int OW, int NPAD, hipStream_t s) {
  int K  = C * KH * KW;
  int K4 = cdiv(K, 4) * 4;
  long long total = (long long)NPAD * K4;
  hipLaunchKernelGGL(k_im2colT, dim3((int)((total + 255) / 256)), dim3(256), 0, s,
                     in, bt, C, H, W, KH, KW, S, OH, OW, NPAD, K, K4);
}

extern "C" void kernel_launch(void* const* d_in, const int* in_sizes, int n_in,
                              void* d_out, int out_size, void* d_ws, size_t ws_size,
                              hipStream_t stream) {
  const float* x     = (const float*)d_in[0];
  const float* z_reg = (const float*)d_in[1];
  const float* z_cls = (const float*)d_in[2];
  const float* bb_w1 = (const float*)d_in[3];  const float* bb_b1 = (const float*)d_in[4];
  const float* bb_w2 = (const float*)d_in[5];  const float* bb_b2 = (const float*)d_in[6];
  const float* bb_w3 = (const float*)d_in[7];  const float* bb_b3 = (const float*)d_in[8];
  const float* bb_w4 = (const float*)d_in[9];  const float* bb_b4 = (const float*)d_in[10];
  const float* bb_w5 = (const float*)d_in[11]; const float* bb_b5 = (const float*)d_in[12];
  const float* wr    = (const float*)d_in[13]; const float* br    = (const float*)d_in[14];
  const float* wc    = (const float*)d_in[15]; const float* bc    = (const float*)d_in[16];
  const float* rh_w1 = (const float*)d_in[17]; const float* rh_b1 = (const float*)d_in[18];
  const float* rh_w2 = (const float*)d_in[19]; const float* rh_b2 = (const float*)d_in[20];
  const float* ch_w1 = (const float*)d_in[21]; const float* ch_b1 = (const float*)d_in[22];
  const float* ch_w2 = (const float*)d_in[23]; const float* ch_b2 = (const float*)d_in[24];
  const float* anc0  = (const float*)d_in[25]; const float* anc1  = (const float*)d_in[26];
  const float* anc2  = (const float*)d_in[27]; const float* anc3  = (const float*)d_in[28];

  float* ws   = (float*)d_ws;
  float* XP   = ws + WS_XP;
  float* APAD = ws + WS_APAD;
  float* A0   = ws + WS_ACT0;
  float* A1   = ws + WS_ACT1;
  float* COL  = ws + WS_COL;

  // 0) NHWC -> NCHW
  hipLaunchKernelGGL(k_nhwc_to_nchw, dim3(cdiv(287 * 287 * 3, 256)), dim3(256), 0, stream,
                     x, XP, 287, 287, 3);

  // 1) conv1 11x11 s2: (3,287,287) -> (96,139,139); relu after pool
  im2colT(XP, COL, 3, 287, 287, 11, 11, 2, 139, 139, pad64(139 * 139), stream);
  gemm(bb_w1, COL, bb_b1, A0, 96, 139 * 139, pad64(139 * 139), 3 * 121, APAD, 0, 0, stream);
  hipLaunchKernelGGL(k_maxpool3s2_relu, dim3(cdiv(96 * 69 * 69, 256)), dim3(256), 0, stream,
                     A0, A1, 96, 139, 139, 69, 69);

  // 2) conv2 5x5: (96,69,69) -> (256,65,65); pool -> (256,32,32) + relu
  im2colT(A1, COL, 96, 69, 69, 5, 5, 1, 65, 65, pad64(65 * 65), stream);
  gemm(bb_w2, COL, bb_b2, A0, 256, 65 * 65, pad64(65 * 65), 96 * 25, APAD, 0, 0, stream);
  hipLaunchKernelGGL(k_maxpool3s2_relu, dim3(cdiv(256 * 32 * 32, 256)), dim3(256), 0, stream,
                     A0, A1, 256, 65, 65, 32, 32);

  // 3) conv3 3x3 relu: (256,32,32) -> (384,30,30)
  im2colT(A1, COL, 256, 32, 32, 3, 3, 1, 30, 30, pad64(900), stream);
  gemm(bb_w3, COL, bb_b3, A0, 384, 900, pad64(900), 256 * 9, APAD, 1, 0, stream);

  // 4) conv4 3x3 relu: (384,30,30) -> (384,28,28)
  im2colT(A0, COL, 384, 30, 30, 3, 3, 1, 28, 28, pad64(784), stream);
  gemm(bb_w4, COL, bb_b4, A1, 384, 784, pad64(784), 384 * 9, APAD, 1, 0, stream);

  // 5) conv5 3x3 (no relu): (384,28,28) -> xf (256,26,26)
  im2colT(A1, COL, 384, 28, 28, 3, 3, 1, 26, 26, pad64(676), stream);
  gemm(bb_w5, COL, bb_b5, A0, 256, 676, pad64(676), 384 * 9, APAD, 0, 0, stream);

  // 6) conv_reg2 (wr) -> c_x ; conv_cls2 (wc) -> r_x ; both relu, share col of xf
  im2colT(A0, COL, 256, 26, 26, 3, 3, 1, 24, 24, pad64(576), stream);
  float* CX = A1;                 // (256,576)
  float* RX = A1 + 147456;        // (256,576)
  gemm(wr, COL, br, CX, 256, 576, pad64(576), 256 * 9, APAD, 1, 0, stream);
  gemm(wc, COL, bc, RX, 256, 576, pad64(576), 256 * 9, APAD, 1, 0, stream);

  // 7) unfold 4x4 (col buffer now dead -> reuse)
  float* CP = ws + WS_CP;
  float* RP = ws + WS_RP;
  hipLaunchKernelGGL(k_unfold44, dim3(cdiv(256 * 16 * 441, 256)), dim3(256), 0, stream, CX, CP);
  hipLaunchKernelGGL(k_unfold44, dim3(cdiv(256 * 16 * 441, 256)), dim3(256), 0, stream, RX, RP);

  // 8) depthwise xcorr, emitted directly in head-GEMM BT layout (3584 x 256)
  float* CCOR = ws + WS_CCOR;     // cls path: c_p (from wr conv) with z_cls
  float* RCOR = ws + WS_RCOR;     // reg path: r_p (from wc conv) with z_reg
  hipLaunchKernelGGL(k_xcorrT, dim3(cdiv(3584 * 256, 256)), dim3(256), 0, stream, CP, z_cls, CCOR);
  hipLaunchKernelGGL(k_xcorrT, dim3(cdiv(3584 * 256, 256)), dim3(256), 0, stream, RP, z_reg, RCOR);

  // 9) heads: 1x1 convs as GEMMs over N = 8*441 = 3528 (NPAD 3584)
  float* H1T  = ws + WS_H1T;
  float* CLST = ws + WS_CLS;
  float* REGT = ws + WS_REG;
  gemm(ch_w1, CCOR, ch_b1, H1T, 256, 3528, 3584, 256, APAD, 1, /*storeT=*/1, stream);
  gemm(ch_w2, H1T, ch_b2, CLST, 10, 3528, 3584, 256, APAD, 0, 0, stream);
  gemm(rh_w1, RCOR, rh_b1, H1T, 256, 3528, 3584, 256, APAD, 1, /*storeT=*/1, stream);
  gemm(rh_w2, H1T, rh_b2, REGT, 20, 3528, 3584, 256, APAD, 0, 0, stream);

  // 10) softmax + anchor decode -> d_out (5 x 8 x 2205)
  hipLaunchKernelGGL(k_decode, dim3(cdiv(8 * 2205, 256)), dim3(256), 0, stream,
                     CLST, REGT, anc0, anc1, anc2, anc3, (float*)d_out);
}